// MixtureOfExperts_70798240907759
// MI455X (gfx1250) — compile-verified
//
#include <hip/hip_runtime.h>
#include <math.h>

#define NTOK   128
#define DIMC   512
#define NEXP   256
#define HDIM   1024
#define TOPK   42
#define MCHUNK 32
#define HT     128
#define XPAD   514   // Xs row stride (floats), breaks LDS bank alignment
#define HPAD   130   // Hs row stride (floats)

typedef float v2f __attribute__((ext_vector_type(2)));
typedef float v8f __attribute__((ext_vector_type(8)));

// ---------------------------------------------------------------------------
// Kernel 1: router — logits, softmax, top-K (deterministic argmax loop),
// second softmax over the selected probabilities (exactly as reference).
// One block per token, 256 threads (one per expert).
// ---------------------------------------------------------------------------
__global__ __launch_bounds__(256) void moe_router(
    const float* __restrict__ x, const float* __restrict__ rw,
    const float* __restrict__ rb, int* __restrict__ top_idx,
    float* __restrict__ top_w)
{
    __shared__ float xrow[DIMC];
    __shared__ float work[NEXP];
    __shared__ float rv[NEXP];
    __shared__ int   ri[NEXP];
    __shared__ float selv[TOPK];
    __shared__ int   seli[TOPK];
    __shared__ float inv_sum;

    const int t   = blockIdx.x;
    const int tid = threadIdx.x;

    xrow[tid]       = x[t * DIMC + tid];
    xrow[tid + 256] = x[t * DIMC + tid + 256];
    __syncthreads();

    float logit = rb[tid];
    #pragma unroll 8
    for (int d = 0; d < DIMC; ++d)
        logit += xrow[d] * rw[d * NEXP + tid];

    // softmax over all experts
    rv[tid] = logit;
    __syncthreads();
    for (int s = 128; s > 0; s >>= 1) {
        if (tid < s) rv[tid] = fmaxf(rv[tid], rv[tid + s]);
        __syncthreads();
    }
    const float mx = rv[0];
    __syncthreads();
    float p = expf(logit - mx);
    rv[tid] = p;
    __syncthreads();
    for (int s = 128; s > 0; s >>= 1) {
        if (tid < s) rv[tid] += rv[tid + s];
        __syncthreads();
    }
    const float prob = p / rv[0];
    __syncthreads();

    work[tid] = prob;
    __syncthreads();

    // iterative argmax: lowest index wins ties (matches jax top_k)
    for (int k = 0; k < TOPK; ++k) {
        rv[tid] = work[tid];
        ri[tid] = tid;
        __syncthreads();
        for (int s = 128; s > 0; s >>= 1) {
            if (tid < s) {
                float v1 = rv[tid + s]; int i1 = ri[tid + s];
                if (v1 > rv[tid] || (v1 == rv[tid] && i1 < ri[tid])) {
                    rv[tid] = v1; ri[tid] = i1;
                }
            }
            __syncthreads();
        }
        if (tid == 0) {
            selv[k] = rv[0];
            seli[k] = ri[0];
            work[ri[0]] = -1.0f;
        }
        __syncthreads();
    }

    // reference re-softmaxes the selected *probabilities*; selv[0] is the max
    if (tid < TOPK) rv[tid] = expf(selv[tid] - selv[0]);
    __syncthreads();
    if (tid == 0) {
        float s = 0.f;
        for (int k = 0; k < TOPK; ++k) s += rv[k];
        inv_sum = 1.0f / s;
    }
    __syncthreads();
    if (tid < TOPK) {
        top_idx[t * TOPK + tid] = seli[tid];
        top_w[t * TOPK + tid]   = rv[tid] * inv_sum;
    }
}

// ---------------------------------------------------------------------------
// Kernel 2: deterministic per-expert token lists via prefix-sum compaction.
// One block per expert. Entry = (token << 8) | slot.
// ---------------------------------------------------------------------------
__global__ __launch_bounds__(256) void moe_build(
    const int* __restrict__ top_idx, int* __restrict__ counts,
    int* __restrict__ lists)
{
    const int e   = blockIdx.x;
    const int tid = threadIdx.x;
    __shared__ int scan[256];
    __shared__ int base;
    if (tid == 0) base = 0;
    __syncthreads();

    const int total = NTOK * TOPK;   // 5376
    for (int c0 = 0; c0 < total; c0 += 256) {
        const int i  = c0 + tid;
        int match = 0, packed = 0;
        if (i < total && top_idx[i] == e) {
            match  = 1;
            packed = ((i / TOPK) << 8) | (i % TOPK);
        }
        scan[tid] = match;
        __syncthreads();
        for (int s = 1; s < 256; s <<= 1) {
            int v = (tid >= s) ? scan[tid - s] : 0;
            __syncthreads();
            scan[tid] += v;
            __syncthreads();
        }
        if (match) lists[e * NTOK + base + scan[tid] - 1] = packed;
        __syncthreads();
        if (tid == 0) base += scan[255];
        __syncthreads();
    }
    if (tid == 0) counts[e] = base;
}

// ---------------------------------------------------------------------------
// Kernel 3: fused expert MLP with f32 WMMA (16x16x4).
// One block (8 waves, 256 thr) per expert; M-chunk 32 tokens; H tiled at 128.
// Wave w: GEMM1 produces Hs cols [w*16,w*16+16); GEMM2 accumulates Y cols
// [w*64,(w+1)*64) in 8 v8f accumulators. Weights stream from HBM exactly
// once per expert chunk. Output to y_partial[(t*K+slot)*DIM] (no atomics).
// ---------------------------------------------------------------------------
__global__ __launch_bounds__(256) void moe_expert(
    const float* __restrict__ w1, const float* __restrict__ b1,
    const float* __restrict__ w2, const float* __restrict__ b2,
    const float* __restrict__ x,  const float* __restrict__ top_w,
    const int* __restrict__ counts, const int* __restrict__ lists,
    float* __restrict__ ypart)
{
    __shared__ __align__(16) float Xs[MCHUNK * XPAD];
    __shared__ __align__(16) float Hs[MCHUNK * HPAD];
    __shared__ int   toks[MCHUNK];
    __shared__ int   slots[MCHUNK];
    __shared__ float wts[MCHUNK];

    const int e = blockIdx.x;
    const int m = counts[e];
    if (m == 0) return;

    const int tid = threadIdx.x;
    const int wv  = tid >> 5;   // wave id 0..7
    const int ln  = tid & 31;
    const int lh  = ln >> 4;    // lane half (selects K pair / M+8)
    const int l15 = ln & 15;

    const float* W1 = w1 + (size_t)e * DIMC * HDIM;
    const float* W2 = w2 + (size_t)e * HDIM * DIMC;
    const float* B1 = b1 + e * HDIM;
    const float* B2 = b2 + e * DIMC;

    for (int mc = 0; mc < m; mc += MCHUNK) {
        int mrem = m - mc; if (mrem > MCHUNK) mrem = MCHUNK;

        if (tid < MCHUNK) {
            if (tid < mrem) {
                const int pk = lists[e * NTOK + mc + tid];
                const int tt = pk >> 8, sl = pk & 255;
                toks[tid]  = tt;
                slots[tid] = sl;
                wts[tid]   = top_w[tt * TOPK + sl];
            } else {
                toks[tid] = 0; slots[tid] = 0; wts[tid] = 0.f;
            }
        }
        __syncthreads();

        // stage X tile (rows beyond mrem zeroed)
        for (int i = tid; i < MCHUNK * DIMC; i += 256) {
            const int r = i >> 9, d = i & (DIMC - 1);
            float v = 0.f;
            if (r < mrem) v = x[toks[r] * DIMC + d];
            Xs[r * XPAD + d] = v;
        }
        __syncthreads();

        v8f acc[2][4];
        #pragma unroll
        for (int mi = 0; mi < 2; ++mi)
            #pragma unroll
            for (int nt = 0; nt < 4; ++nt)
                acc[mi][nt] = (v8f){0.f,0.f,0.f,0.f,0.f,0.f,0.f,0.f};

        for (int ht = 0; ht < HDIM; ht += HT) {
            // ---- GEMM1: h = X(32x512) @ W1[:, hb:hb+16], this wave's 16 cols
            const int hb = ht + wv * 16;
            v8f c0 = (v8f){0.f,0.f,0.f,0.f,0.f,0.f,0.f,0.f};
            v8f c1 = (v8f){0.f,0.f,0.f,0.f,0.f,0.f,0.f,0.f};
            const float* w1col = W1 + hb + l15;
            #pragma unroll 4
            for (int k0 = 0; k0 < DIMC; k0 += 4) {
                const int kk = k0 + 2 * lh;
                v2f a0 = *(const v2f*)&Xs[l15 * XPAD + kk];
                v2f a1 = *(const v2f*)&Xs[(l15 + 16) * XPAD + kk];
                v2f b;
                b.x = w1col[(size_t)kk * HDIM];
                b.y = w1col[(size_t)(kk + 1) * HDIM];
                c0 = __builtin_amdgcn_wmma_f32_16x16x4_f32(false, a0, false, b,
                        (short)0, c0, false, false);
                c1 = __builtin_amdgcn_wmma_f32_16x16x4_f32(false, a1, false, b,
                        (short)0, c1, false, false);
            }
            // bias + exact-erf GELU, store to Hs
            const float bb = B1[hb + l15];
            const int   hc = wv * 16 + l15;
            #pragma unroll
            for (int i = 0; i < 8; ++i) {
                const int mr = i + 8 * lh;
                float v0 = c0[i] + bb;
                float v1 = c1[i] + bb;
                v0 = 0.5f * v0 * (1.0f + erff(v0 * 0.70710678118654752f));
                v1 = 0.5f * v1 * (1.0f + erff(v1 * 0.70710678118654752f));
                Hs[mr * HPAD + hc]        = v0;
                Hs[(mr + 16) * HPAD + hc] = v1;
            }
            __syncthreads();

            // ---- GEMM2: Y[:, wave's 64 cols] += Hs(32xHT) @ W2[ht:ht+HT, :]
            #pragma unroll 4
            for (int k0 = 0; k0 < HT; k0 += 4) {
                const int kk = k0 + 2 * lh;
                v2f a0 = *(const v2f*)&Hs[l15 * HPAD + kk];
                v2f a1 = *(const v2f*)&Hs[(l15 + 16) * HPAD + kk];
                const float* w2row = W2 + (size_t)(ht + kk) * DIMC + l15;
                #pragma unroll
                for (int nt = 0; nt < 4; ++nt) {
                    const int cb = wv * 64 + nt * 16;
                    v2f b;
                    b.x = w2row[cb];
                    b.y = w2row[cb + DIMC];
                    acc[0][nt] = __builtin_amdgcn_wmma_f32_16x16x4_f32(false, a0,
                            false, b, (short)0, acc[0][nt], false, false);
                    acc[1][nt] = __builtin_amdgcn_wmma_f32_16x16x4_f32(false, a1,
                            false, b, (short)0, acc[1][nt], false, false);
                }
            }
            __syncthreads();
        }

        // ---- epilogue: ypart[(t*K+slot)*DIM + col] = (acc + b2[col]) * wt
        #pragma unroll
        for (int mi = 0; mi < 2; ++mi) {
            #pragma unroll
            for (int nt = 0; nt < 4; ++nt) {
                const int cb  = wv * 64 + nt * 16 + l15;
                const float bv = B2[cb];
                #pragma unroll
                for (int i = 0; i < 8; ++i) {
                    const int mr = mi * 16 + i + 8 * lh;
                    if (mr < mrem) {
                        const float v = (acc[mi][nt][i] + bv) * wts[mr];
                        ypart[((size_t)toks[mr] * TOPK + slots[mr]) * DIMC + cb] = v;
                    }
                }
            }
        }
        __syncthreads();
    }
}

// ---------------------------------------------------------------------------
// Kernel 4: fixed-order reduction over the K slots per token (deterministic).
// ---------------------------------------------------------------------------
__global__ __launch_bounds__(256) void moe_reduce(
    const float* __restrict__ ypart, float* __restrict__ out)
{
    const int i = blockIdx.x * 256 + threadIdx.x;   // over NTOK*DIMC
    const int t = i >> 9, c = i & (DIMC - 1);
    const float* p = ypart + (size_t)t * TOPK * DIMC + c;
    float s = 0.f;
    #pragma unroll 6
    for (int k = 0; k < TOPK; ++k) s += p[k * DIMC];
    out[i] = s;
}

// ---------------------------------------------------------------------------
extern "C" void kernel_launch(void* const* d_in, const int* in_sizes, int n_in,
                              void* d_out, int out_size, void* d_ws, size_t ws_size,
                              hipStream_t stream)
{
    const float* x  = (const float*)d_in[0];
    const float* w1 = (const float*)d_in[1];
    const float* b1 = (const float*)d_in[2];
    const float* w2 = (const float*)d_in[3];
    const float* b2 = (const float*)d_in[4];
    const float* rw = (const float*)d_in[5];
    const float* rb = (const float*)d_in[6];
    float* out = (float*)d_out;

    char* ws = (char*)d_ws;
    int*   top_idx = (int*)  (ws + 0);        // N*K ints   = 21504 B
    float* top_w   = (float*)(ws + 21760);    // N*K floats = 21504 B
    int*   counts  = (int*)  (ws + 43520);    // E ints     = 1024 B
    int*   lists   = (int*)  (ws + 44544);    // E*128 ints = 131072 B
    float* ypart   = (float*)(ws + 175616);   // N*K*DIM f32 = 11010048 B

    moe_router<<<NTOK, 256, 0, stream>>>(x, rw, rb, top_idx, top_w);
    moe_build <<<NEXP, 256, 0, stream>>>(top_idx, counts, lists);
    moe_expert<<<NEXP, 256, 0, stream>>>(w1, b1, w2, b2, x, top_w,
                                         counts, lists, ypart);
    moe_reduce<<<(NTOK * DIMC) / 256, 256, 0, stream>>>(ypart, out);
}